// SetCriterion_67422396613236
// MI455X (gfx1250) — compile-verified
//
#include <hip/hip_runtime.h>
#include <math.h>

typedef __attribute__((ext_vector_type(2))) float v2f;
typedef __attribute__((ext_vector_type(8))) float v8f;

#define EPSF 1e-8f

// ---------------------------------------------------------------------------
// Block-wide sum reduction (256 threads). Returns total to all threads.
// ---------------------------------------------------------------------------
__device__ __forceinline__ float blockReduceSum256(float v) {
    __shared__ float s[256];
    int t = threadIdx.x;
    __syncthreads();               // protect against reuse across calls
    s[t] = v;
    __syncthreads();
    for (int o = 128; o > 0; o >>= 1) {
        if (t < o) s[t] += s[t + o];
        __syncthreads();
    }
    return s[0];
}

// ---------------------------------------------------------------------------
// as_loss partials: sum over (b,h,w) of (sum_t sep[t,hw] - mix[b,hw])^2
// N = 512*1024 = 524288 spatial positions; 20 sep slices, 20 mix slices.
// ---------------------------------------------------------------------------
__global__ __launch_bounds__(256) void as_kernel(const float* __restrict__ sep,
                                                 const float* __restrict__ mix,
                                                 float* __restrict__ part) {
    const long N = 524288;
    float acc = 0.f;
    for (long idx = (long)blockIdx.x * blockDim.x + threadIdx.x; idx < N;
         idx += (long)gridDim.x * blockDim.x) {
        float s = 0.f;
        for (int t = 0; t < 20; ++t) s += sep[(long)t * N + idx];
        for (int b = 0; b < 20; ++b) {
            float d = s - mix[(long)b * N + idx];
            acc += d * d;
        }
    }
    float tot = blockReduceSum256(acc);
    if (threadIdx.x == 0) part[blockIdx.x] = tot;
}

// ---------------------------------------------------------------------------
// Spectral flatness per row of 1024: r = exp(mean(log max(1e-4,s^2)))/mean(...)
// Also r*log1p(mean(s^2)). 8192 rows, one block per row.
// ---------------------------------------------------------------------------
__global__ __launch_bounds__(256) void sf_kernel(const float* __restrict__ S,
                                                 float* __restrict__ r_arr,
                                                 float* __restrict__ rl_arr) {
    long row = blockIdx.x;                 // 0..8191  (b*4+i)*512 + t
    const float* p = S + row * 1024;
    float sl = 0.f, sx = 0.f, s2 = 0.f;
    for (int k = threadIdx.x; k < 1024; k += 256) {
        float v  = p[k];
        float v2 = v * v;
        float xc = fmaxf(1e-4f, v2);
        sl += logf(xc);
        sx += xc;
        s2 += v2;
    }
    sl = blockReduceSum256(sl);
    sx = blockReduceSum256(sx);
    s2 = blockReduceSum256(s2);
    if (threadIdx.x == 0) {
        float g     = expf(sl * (1.0f / 1024.0f));
        float a     = sx * (1.0f / 1024.0f);
        float r     = g / a;
        float power = s2 * (1.0f / 1024.0f);
        r_arr[row]  = r;
        rl_arr[row] = r * log1pf(power);
    }
}

// ---------------------------------------------------------------------------
// sm_loss partials: pred_masks (4,4,256,256); over (b,u,v): -log(max_i/sum_i)
// ---------------------------------------------------------------------------
__global__ __launch_bounds__(256) void sm_kernel(const float* __restrict__ pm,
                                                 float* __restrict__ part) {
    const int N = 4 * 65536;
    float acc = 0.f;
    for (int p = blockIdx.x * blockDim.x + threadIdx.x; p < N;
         p += gridDim.x * blockDim.x) {
        int  b    = p >> 16;
        int  uv   = p & 65535;
        long base = (long)b * 262144 + uv;
        float m = -3.4e38f, s = 0.f;
        for (int i = 0; i < 4; ++i) {
            float v = pm[base + (long)i * 65536];
            m = fmaxf(m, v);
            s += v;
        }
        acc += logf(s) - logf(m);          // = -log(m/s)
    }
    float tot = blockReduceSum256(acc);
    if (threadIdx.x == 0) part[blockIdx.x] = tot;
}

// ---------------------------------------------------------------------------
// P[i,c] = sum over 256*256 contiguous floats of pp_embeds[i,c,:,:]
// One block per (i,c) pair; float4-vectorized streaming reduction (268 MB —
// the dominant bandwidth cost of the whole problem).
// ---------------------------------------------------------------------------
__global__ __launch_bounds__(256) void psum_kernel(const float* __restrict__ pp,
                                                   float* __restrict__ P) {
    long base = (long)blockIdx.x * 65536;
    const float4* p4 = (const float4*)(pp + base);
    float acc = 0.f;
    #pragma unroll 4
    for (int it = 0; it < 64; ++it) {
        float4 v = p4[it * 256 + threadIdx.x];
        acc += v.x + v.y + v.z + v.w;
    }
    float tot = blockReduceSum256(acc);
    if (threadIdx.x == 0) P[blockIdx.x] = tot;
}

// ---------------------------------------------------------------------------
// _diff_loss via WMMA: one wave32 computes the 16x16 gram G = nv * nv^T of the
// 16 row-normalized 256-dim vectors with V_WMMA_F32_16X16X4_F32 (K=256 in 64
// steps of K=4), then extracts the per-b upper-triangular blocks.
//
// A layout (16x4 f32, ISA 7.12.2): lane L, VGPR v  ->  A[L%16][2*(L/16)+v]
// B layout is the row-striped transpose -> same lane data serves as B for the
// gram product.  C/D: lane n, VGPR v -> G[v + 8*(lane/16)][lane%16].
// Row base = (r/4)*bStride + (r%4)*iStride (handles the x[:4,:4] slice of the
// (16,512,256) features tensor and the dense (4,4,256) tokens tensor).
// ---------------------------------------------------------------------------
__global__ __launch_bounds__(32) void gram_diff_kernel(const float* __restrict__ x,
                                                       int bStride, int iStride,
                                                       float* __restrict__ outSlot) {
    int lane  = threadIdx.x;            // 0..31, EXEC all ones
    int row   = lane & 15;              // M index this lane feeds
    int koff  = (lane >> 4) * 2;        // K sub-offset: 0 or 2
    long base = (long)(row >> 2) * bStride + (long)(row & 3) * iStride;

    // Row 2-norm (redundant across lane halves -> uniform control flow).
    float nrm2 = 0.f;
    for (int k = 0; k < 256; ++k) {
        float v = x[base + k];
        nrm2 += v * v;
    }
    float inv = 1.0f / fmaxf(sqrtf(nrm2), EPSF);

    v8f c = {};
    for (int k0 = 0; k0 < 256; k0 += 4) {
        v2f a;
        a.x = x[base + k0 + koff]     * inv;
        a.y = x[base + k0 + koff + 1] * inv;
        // D = A * A^T + C  (16x16x4 f32 WMMA, accumulate over K)
        c = __builtin_amdgcn_wmma_f32_16x16x4_f32(
                /*neg_a=*/false, a, /*neg_b=*/false, a,
                /*c_mod=*/(short)0, c, /*reuse_a=*/false, /*reuse_b=*/false);
    }

    // Masked extraction: same 4-block (m/4==n/4), strict upper (m%4 < n%4).
    float acc  = 0.f;
    int n      = lane & 15;
    int mbase  = (lane >> 4) * 8;
    for (int v = 0; v < 8; ++v) {
        int m = mbase + v;
        if ((m >> 2) == (n >> 2) && (m & 3) < (n & 3))
            acc += 1.0f - fabsf(c[v]) + 1e-4f;
    }

    __shared__ float red[32];
    red[lane] = acc;
    __syncthreads();
    if (lane == 0) {
        float t = 0.f;
        for (int i = 0; i < 32; ++i) t += red[i];
        *outSlot = t;
    }
}

// ---------------------------------------------------------------------------
// Finalize: reduce all partials, compute fl/ad, 4x4 clip logits + CE, write 7.
// ---------------------------------------------------------------------------
__global__ __launch_bounds__(256) void final_kernel(
        const float* __restrict__ asPart, int nAs,
        const float* __restrict__ smPart, int nSm,
        const float* __restrict__ r_arr,  const float* __restrict__ rl_arr,
        const float* __restrict__ P,      const float* __restrict__ gramOut,
        const float* __restrict__ tokens, const float* __restrict__ comp,
        const float* __restrict__ embeds, float* __restrict__ out) {
    int t = threadIdx.x;

    float a = 0.f;
    for (int i = t; i < nAs; i += 256) a += asPart[i];
    float asSum = blockReduceSum256(a);

    float b = 0.f;
    for (int i = t; i < nSm; i += 256) b += smPart[i];
    float smSum = blockReduceSum256(b);

    float c = 0.f;
    for (int i = t; i < 8192; i += 256) c += rl_arr[i];
    float lsSum = blockReduceSum256(c);

    __shared__ float fl_s[16];
    if (t < 16) {
        float s = 0.f;
        for (int k = 0; k < 512; ++k) s += r_arr[t * 512 + k];
        fl_s[t] = s * (1.0f / 512.0f);
    }
    __syncthreads();

    // ad_loss: fl[b,i] * |cos(tokens[b,i], comp[b,i])|
    float adv = 0.f;
    if (t < 16) {
        const float* u = tokens + t * 256;
        const float* v = comp   + t * 256;
        float d = 0.f, na = 0.f, nb = 0.f;
        for (int k = 0; k < 256; ++k) {
            d  += u[k] * v[k];
            na += u[k] * u[k];
            nb += v[k] * v[k];
        }
        float cosv = d / (fmaxf(sqrtf(na), EPSF) * fmaxf(sqrtf(nb), EPSF));
        adv = fl_s[t] * fabsf(cosv);
    }
    float adSum = blockReduceSum256(adv);

    // E[j,c] = sum_i embeds[j,i,c]
    __shared__ float E_s[1024];
    for (int q = t; q < 1024; q += 256) {
        int j = q >> 8, cc = q & 255;
        float s = 0.f;
        for (int i = 0; i < 4; ++i) s += embeds[j * 1024 + i * 256 + cc];
        E_s[q] = s;
    }
    __syncthreads();

    __shared__ float L_s[16];
    if (t < 16) {
        int i = t >> 2, j = t & 3;
        float s = 0.f;
        for (int cc = 0; cc < 256; ++cc) s += P[i * 256 + cc] * E_s[j * 256 + cc];
        L_s[t] = s;
    }
    __syncthreads();

    if (t == 0) {
        float L[4][4];
        for (int i = 0; i < 4; ++i) {
            float den = 0.f;
            for (int j = 0; j < 4; ++j) den += fabsf(L_s[i * 4 + j]);
            for (int j = 0; j < 4; ++j) L[i][j] = L_s[i * 4 + j] / den;
        }
        float ce1 = 0.f, ce2 = 0.f;
        for (int i = 0; i < 4; ++i) {
            float mx = L[i][0];
            for (int j = 1; j < 4; ++j) mx = fmaxf(mx, L[i][j]);
            float se = 0.f;
            for (int j = 0; j < 4; ++j) se += expf(L[i][j] - mx);
            ce1 += -(L[i][i] - mx - logf(se));

            float mxc = L[0][i];
            for (int j = 1; j < 4; ++j) mxc = fmaxf(mxc, L[j][i]);
            float sec = 0.f;
            for (int j = 0; j < 4; ++j) sec += expf(L[j][i] - mxc);
            ce2 += -(L[i][i] - mxc - logf(sec));
        }
        ce1 *= 0.25f;
        ce2 *= 0.25f;

        out[0] = sqrtf(asSum);      // as_loss
        out[1] = lsSum;             // ls_loss
        out[2] = gramOut[0];        // df_loss (WMMA)
        out[3] = gramOut[1];        // dt_loss (WMMA)
        out[4] = adSum;             // ad_loss
        out[5] = smSum;             // sm_loss
        out[6] = 0.5f * (ce1 + ce2);// clip_loss
    }
}

// ---------------------------------------------------------------------------
extern "C" void kernel_launch(void* const* d_in, const int* in_sizes, int n_in,
                              void* d_out, int out_size, void* d_ws, size_t ws_size,
                              hipStream_t stream) {
    const float* mixed  = (const float*)d_in[0]; // (20,1,512,1024)
    const float* seps   = (const float*)d_in[1]; // (20,1,512,1024)
    const float* wo     = (const float*)d_in[2]; // (4,4,512,1024)
    const float* feats  = (const float*)d_in[3]; // (16,512,256)
    const float* tokens = (const float*)d_in[4]; // (4,4,256)
    const float* comp   = (const float*)d_in[5]; // (4,4,256)
    const float* pm     = (const float*)d_in[6]; // (4,4,256,256)
    const float* embeds = (const float*)d_in[7]; // (4,4,256)
    const float* pp     = (const float*)d_in[8]; // (4,256,256,256)
    float* out = (float*)d_out;

    float* ws     = (float*)d_ws;
    float* asPart = ws;             // 512
    float* r_arr  = ws + 512;       // 8192
    float* rl_arr = ws + 8704;      // 8192
    float* smPart = ws + 16896;     // 256
    float* P      = ws + 17152;     // 1024
    float* gramO  = ws + 18176;     // 2

    as_kernel   <<<512, 256, 0, stream>>>(seps, mixed, asPart);
    sf_kernel   <<<8192, 256, 0, stream>>>(wo, r_arr, rl_arr);
    sm_kernel   <<<256, 256, 0, stream>>>(pm, smPart);
    psum_kernel <<<1024, 256, 0, stream>>>(pp, P);
    // df: rows come from feats[:4,:4] of (16,512,256) -> bStride=512*256, iStride=256
    gram_diff_kernel<<<1, 32, 0, stream>>>(feats, 512 * 256, 256, gramO + 0);
    // dt: tokens (4,4,256) dense -> bStride=1024, iStride=256
    gram_diff_kernel<<<1, 32, 0, stream>>>(tokens, 1024, 256, gramO + 1);
    final_kernel<<<1, 256, 0, stream>>>(asPart, 512, smPart, 256, r_arr, rl_arr,
                                        P, gramO, tokens, comp, embeds, out);
}